// Ga3Conv2d_38096359915961
// MI455X (gfx1250) — compile-verified
//
#include <hip/hip_runtime.h>

typedef __attribute__((ext_vector_type(2))) float v2f;
typedef __attribute__((ext_vector_type(8))) float v8f;

// For each (m, k): the unique (j, sign) with S[m,j,k] != 0, transcribed from _TERMS.
__device__ const signed char J_OF[8][8] = {
    {0,1,2,3,4,5,6,7},
    {1,0,4,6,2,7,3,5},
    {2,4,0,5,1,3,7,6},
    {3,6,5,0,7,2,1,4},
    {4,2,1,7,0,6,5,3},
    {5,7,3,2,6,0,4,1},
    {6,3,7,1,5,4,0,2},
    {7,5,6,4,3,1,2,0},
};
__device__ const signed char SGN[8][8] = {
    {+1,+1,+1,+1,-1,-1,-1,-1},
    {+1,+1,-1,-1,+1,-1,+1,-1},
    {+1,+1,+1,-1,-1,+1,+1,+1},
    {+1,+1,+1,+1,-1,-1,-1,-1},
    {+1,+1,-1,+1,+1,+1,-1,+1},
    {+1,+1,+1,-1,-1,+1,+1,+1},
    {+1,+1,-1,-1,+1,-1,+1,-1},
    {+1,+1,-1,+1,+1,+1,-1,+1},
};

// ---------------------------------------------------------------------------
// Stage 0: fold the GA sign table into the conv weights, emitting them in the
// exact V_WMMA_F32_16X16X4_F32 A-fragment lane layout:
//   wsA[((t*288 + kkc)*32 + lane)*2 + e] = Wc[oc = t*16 + (lane&15)]
//                                            [kk = kkc*4 + (lane>>4)*2 + e]
// with kk = tap*128 + ic, ic = cin*8 + k_blade, oc = cout*8 + m.
// Also folds the bias: wsBias[oc] = sum_k sign(m,k) * b[j(m,k), cout].
// ---------------------------------------------------------------------------
__global__ void ga3_build_w(const float* __restrict__ W,
                            const float* __restrict__ bias,
                            float* __restrict__ wsA,
                            float* __restrict__ wsBias) {
  int gid = blockIdx.x * 256 + threadIdx.x;
  if (gid < 8 * 288 * 32 * 2) {          // 147456 A-fragment elements
    int e    = gid & 1;
    int lane = (gid >> 1) & 31;
    int rest = gid >> 6;
    int kkc  = rest % 288;
    int t    = rest / 288;
    int kk   = kkc * 4 + (lane >> 4) * 2 + e;   // 0..1151
    int tap  = kk >> 7;                          // 0..8  (dy*3+dx)
    int ic   = kk & 127;                         // input channel
    int kb   = ic & 7;                           // blade of input channel
    int cin  = ic >> 3;
    int oc   = t * 16 + (lane & 15);             // output channel
    int m    = oc & 7;
    int cout = oc >> 3;
    int j    = J_OF[m][kb];
    float s  = (float)SGN[m][kb];
    // W layout: [8, 16, 16, 3, 3] -> ((j*16+cout)*16+cin)*9 + tap
    wsA[gid] = s * W[((j * 16 + cout) * 16 + cin) * 9 + tap];
  }
  if (gid < 128) {
    int m = gid & 7, cout = gid >> 3;
    float acc = 0.0f;
    #pragma unroll
    for (int k = 0; k < 8; ++k)
      acc += (float)SGN[m][k] * bias[J_OF[m][k] * 16 + cout];
    wsBias[gid] = acc;
  }
}

// ---------------------------------------------------------------------------
// Stage 1: implicit-GEMM 3x3 conv via V_WMMA_F32_16X16X4_F32.
//   Block = 256 threads (8 wave32). Block tile: 128 out-channels x 64 pixels.
//   Wave w: out-channels [w*16, w*16+16) x 64 px = four 16x16 C tiles, so each
//   A-fragment load (global_load_b64, L2-hot) feeds 4 WMMAs.
//   K = 9 taps * 128 channels = 1152 -> 288 K-steps, 1152 WMMA per wave.
//   Input halo (3 rows x 66 cols x 128 ch) staged in dynamic LDS (~102 KB,
//   col stride padded to 68 for bank spread); 3 blocks/WGP on 320 KB LDS.
// ---------------------------------------------------------------------------
__global__ __launch_bounds__(256)
void ga3_conv_wmma(const float* __restrict__ x,
                   const float* __restrict__ wsA,
                   const float* __restrict__ wsBias,
                   float* __restrict__ out) {
  extern __shared__ float lds[];   // [ic(128)][dy(3)][col: stride 68, 66 used]

  int tile = blockIdx.x;           // 2 w-tiles * 128 rows * 8 images = 2048
  int w0   = (tile & 1) << 6;
  int h    = (tile >> 1) & 127;
  int bimg = tile >> 8;

  const float* xb = x + (size_t)bimg * (128 * 128 * 128);

  // ---- stage input halo into LDS (zero-padded at image borders) ----
  for (int i = threadIdx.x; i < 128 * 3 * 66; i += 256) {
    int ic  = i / 198;
    int r   = i - ic * 198;
    int dy  = r / 66;
    int col = r - dy * 66;
    int gh  = h + dy - 1;
    int gw  = w0 + col - 1;
    float v = 0.0f;
    if ((unsigned)gh < 128u && (unsigned)gw < 128u)
      v = xb[(ic * 128 + gh) * 128 + gw];
    lds[ic * 204 + dy * 68 + col] = v;
  }
  __syncthreads();

  int wave = threadIdx.x >> 5;
  int lane = threadIdx.x & 31;
  int n    = lane & 15;            // pixel-in-subtile / M-row of fragment
  int hi   = lane >> 4;            // K half-select (K0/K1 vs K2/K3)

  v8f acc[4] = {{}, {}, {}, {}};
  const float* Ap = wsA + (size_t)wave * (288 * 64) + lane * 2;

  #pragma unroll
  for (int tap = 0; tap < 9; ++tap) {
    const int dy = tap / 3, dx = tap - dy * 3;
    const int base = hi * 2 * 204 + dy * 68 + dx + n;  // channel hi*2, row dy
    #pragma unroll 2
    for (int icc = 0; icc < 32; ++icc) {
      // A fragment: 16x4 slab of folded weights, one b64 per lane, reused 4x.
      v2f a = *(const v2f*)(Ap + (size_t)(tap * 32 + icc) * 64);
      const int cb = base + icc * 816;                 // + icc*4 channels
      #pragma unroll
      for (int t = 0; t < 4; ++t) {
        v2f bf;                                        // 4x16 input slab
        bf.x = lds[cb + t * 16];                       // K even (ch +0)
        bf.y = lds[cb + t * 16 + 204];                 // K odd  (ch +1)
        acc[t] = __builtin_amdgcn_wmma_f32_16x16x4_f32(
            /*neg_a=*/false, a, /*neg_b=*/false, bf,
            /*c_mod=*/(short)0, acc[t], /*reuse_a=*/false, /*reuse_b=*/false);
      }
    }
  }

  // ---- bias + store: C VGPR v holds row M = v + hi*8, col = n ----
  float bvals[8];
  #pragma unroll
  for (int v = 0; v < 8; ++v) bvals[v] = wsBias[wave * 16 + v + hi * 8];

  float* ob = out + (size_t)bimg * (128 * 128 * 128) + (size_t)h * 128 + w0 + n;
  #pragma unroll
  for (int t = 0; t < 4; ++t) {
    #pragma unroll
    for (int v = 0; v < 8; ++v) {
      int oc = wave * 16 + v + hi * 8;
      ob[(size_t)oc * 16384 + t * 16] = acc[t][v] + bvals[v];
    }
  }
}

extern "C" void kernel_launch(void* const* d_in, const int* in_sizes, int n_in,
                              void* d_out, int out_size, void* d_ws, size_t ws_size,
                              hipStream_t stream) {
  const float* x = (const float*)d_in[0];   // [8, 128, 128, 128]
  const float* W = (const float*)d_in[1];   // [8, 16, 16, 3, 3]
  const float* b = (const float*)d_in[2];   // [8, 16]
  float* out    = (float*)d_out;            // [8, 128, 128, 128]
  float* wsA    = (float*)d_ws;             // 147456 floats (A fragments)
  float* wsBias = wsA + 147456;             // 128 floats

  ga3_build_w<<<576, 256, 0, stream>>>(W, b, wsA, wsBias);
  // dynamic LDS: 128 ch * 3 rows * 68 col-stride * 4 B = 104448 bytes
  ga3_conv_wmma<<<2048, 256, 104448, stream>>>(x, wsA, wsBias, out);
}